// RGCN_7533372637993
// MI455X (gfx1250) — compile-verified
//
#include <hip/hip_runtime.h>

#define N_NODES 100000
#define N_EDGES 1600000
#define DD 128
#define RR 8
#define BB 4
#define LL 2
#define OUTC 40

typedef __attribute__((ext_vector_type(2))) float v2f;
typedef __attribute__((ext_vector_type(8))) float v8f;

// W[r,i,o] = sum_b comb[r,b] * V[b,i,o]
__global__ void build_w_kernel(const float* __restrict__ V,    // [B,D,D]
                               const float* __restrict__ comb, // [R,B]
                               float* __restrict__ W) {        // [R,D,D]
  int idx = blockIdx.x * blockDim.x + threadIdx.x;
  if (idx >= RR * DD * DD) return;
  int r = idx / (DD * DD);
  int io = idx - r * (DD * DD);
  float acc = 0.f;
#pragma unroll
  for (int bb = 0; bb < BB; ++bb)
    acc = fmaf(comb[r * BB + bb], V[(size_t)bb * DD * DD + io], acc);
  W[idx] = acc;
}

// C[M x NC] = A[M x 128] @ Bm[128 x NC] (+ bias), M % 16 == 0, K = 128.
// grid.x = M/16, blockDim.x = 32 * ceil(NC/16); wave w owns column tile w.
// NC is a compile-time constant so all B-row strides fold into the 24-bit
// instruction offset field (no per-iteration 64-bit VALU address adds).
// fp32 WMMA 16x16x4 layout:
//   A: lanes 0-15 -> rows m0+lr, K = kk+{0,1}; lanes 16-31 -> K = kk+{2,3}
//   B: v0 = {K=kk (lanes0-15), K=kk+2 (lanes16-31)}, v1 = {K=kk+1, K=kk+3}
//   C: VGPR v -> row m0+v (lanes0-15) / m0+8+v (lanes16-31), col n0+lr
template <int NC>
__global__ void wmma_gemm128_kernel(const float* __restrict__ A,
                                    const float* __restrict__ Bm,
                                    const float* __restrict__ bias,
                                    float* __restrict__ C) {
  constexpr bool GUARD = (NC % 16) != 0;
  const int lane = threadIdx.x & 31;
  const int wave = threadIdx.x >> 5;
  const int m0 = blockIdx.x << 4;
  const int n0 = wave << 4;
  const int half = lane >> 4;
  const int lr = lane & 15;
  const int col = n0 + lr;
  const bool colok = GUARD ? (col < NC) : true;

  const float* arow = A + (size_t)(m0 + lr) * DD + 2 * half;
  const float* bcol = Bm + (size_t)(2 * half) * NC + (colok ? col : 0);

  v8f acc = {};
#pragma unroll
  for (int kk = 0; kk < DD; kk += 4) {
    v2f a, b;
    a.x = arow[kk];
    a.y = arow[kk + 1];
    float bx = bcol[kk * NC];
    float by = bcol[(kk + 1) * NC];
    if constexpr (GUARD) {
      b.x = colok ? bx : 0.f;
      b.y = colok ? by : 0.f;
    } else {
      b.x = bx;
      b.y = by;
    }
    acc = __builtin_amdgcn_wmma_f32_16x16x4_f32(false, a, false, b, (short)0,
                                                acc, false, false);
  }

  if (colok) {
    float bv = bias ? bias[col] : 0.f;
#pragma unroll
    for (int v = 0; v < 8; ++v) {
      int row = m0 + v + 8 * half;
      C[(size_t)row * NC + col] = acc[v] + bv;
    }
  }
}

// One wave per edge; edges of relation `rel` only:
//   agg[dst[e], :] += proj[src[e], :]   (128 floats, float4 per lane)
__global__ void scatter_kernel(const float* __restrict__ proj,
                               const int* __restrict__ src,
                               const int* __restrict__ dst,
                               const int* __restrict__ et, int rel,
                               float* __restrict__ agg) {
  int e = blockIdx.x * (blockDim.x >> 5) + (threadIdx.x >> 5);
  if (e >= N_EDGES) return;
  if (et[e] != rel) return;
  int lane = threadIdx.x & 31;
  const float4* p = (const float4*)(proj + (size_t)src[e] * DD);
  float4 v = p[lane];
  float* a = agg + (size_t)dst[e] * DD + (size_t)lane * 4;
  atomicAdd(a + 0, v.x);
  atomicAdd(a + 1, v.y);
  atomicAdd(a + 2, v.z);
  atomicAdd(a + 3, v.w);
}

__global__ void relu_kernel(const float* __restrict__ in,
                            float* __restrict__ out, size_t n) {
  size_t i = (size_t)blockIdx.x * blockDim.x + threadIdx.x;
  if (i < n) out[i] = fmaxf(in[i], 0.f);
}

extern "C" void kernel_launch(void* const* d_in, const int* in_sizes, int n_in,
                              void* d_out, int out_size, void* d_ws,
                              size_t ws_size, hipStream_t stream) {
  (void)in_sizes; (void)n_in; (void)out_size; (void)ws_size;
  const float* x     = (const float*)d_in[0];  // [N,D]
  const int*   src   = (const int*)d_in[1];    // [E]
  const int*   dst   = (const int*)d_in[2];    // [E]
  const int*   et    = (const int*)d_in[3];    // [E]
  const float* V     = (const float*)d_in[4];  // [L,B,D,D]
  const float* comb  = (const float*)d_in[5];  // [L,R,B]
  const float* Wself = (const float*)d_in[6];  // [L,D,D]
  const float* bvec  = (const float*)d_in[7];  // [L,D]
  const float* Wout  = (const float*)d_in[8];  // [D,OUT]
  const float* bout  = (const float*)d_in[9];  // [OUT]

  float* out    = (float*)d_out;                       // [N,OUT] then [N,D]
  float* hfinal = out + (size_t)N_NODES * OUTC;

  // workspace layout (floats): W[R*D*D] | proj[N*D] | agg[N*D] | h1[N*D]
  float* Wbuf = (float*)d_ws;
  float* proj = Wbuf + (size_t)RR * DD * DD;
  float* agg  = proj + (size_t)N_NODES * DD;
  float* h1   = agg + (size_t)N_NODES * DD;

  const float* hin = x;
  for (int l = 0; l < LL; ++l) {
    build_w_kernel<<<(RR * DD * DD + 255) / 256, 256, 0, stream>>>(
        V + (size_t)l * BB * DD * DD, comb + (size_t)l * RR * BB, Wbuf);

    // agg = hin @ Wself[l] + b[l]   (seeds the edge accumulator)
    wmma_gemm128_kernel<DD><<<N_NODES / 16, 256, 0, stream>>>(
        hin, Wself + (size_t)l * DD * DD, bvec + (size_t)l * DD, agg);

    for (int r = 0; r < RR; ++r) {
      // proj = hin @ W[r]
      wmma_gemm128_kernel<DD><<<N_NODES / 16, 256, 0, stream>>>(
          hin, Wbuf + (size_t)r * DD * DD, nullptr, proj);
      // agg[dst] += proj[src] for edges of relation r
      scatter_kernel<<<N_EDGES / 8, 256, 0, stream>>>(proj, src, dst, et, r,
                                                      agg);
    }

    float* hout = (l == LL - 1) ? hfinal : h1;
    size_t n = (size_t)N_NODES * DD;
    relu_kernel<<<(unsigned)((n + 255) / 256), 256, 0, stream>>>(agg, hout, n);
    hin = hout;
  }

  // out = h @ Wout + bout  (NC = 40 -> 3 column tiles, last tile guarded)
  wmma_gemm128_kernel<OUTC><<<N_NODES / 16, 96, 0, stream>>>(hin, Wout, bout,
                                                             out);
}